// PointTransformer_51196010169068
// MI455X (gfx1250) — compile-verified
//
#include <hip/hip_runtime.h>
#include <math.h>

// Problem constants (match reference)
#define BD   8
#define PPTS 4096
#define NPTS (BD * PPTS)   // 32768 points total
#define KNN  16

typedef __attribute__((ext_vector_type(2))) float v2f;
typedef __attribute__((ext_vector_type(8))) float v8f;

__device__ __forceinline__ float wave_sum32(float s) {
#pragma unroll
  for (int off = 16; off > 0; off >>= 1) s += __shfl_xor(s, off, 32);
  return s;
}

// ---------------------------------------------------------------------------
// KNN: one wave32 per query point. Per-lane register-resident sorted top-16,
// then LDS-staged 32-way merge with shfl_xor min-reductions.
// d2 = |p|^2 + |q|^2 - 2 p.q  (same formula as reference), self masked +1e10.
// ---------------------------------------------------------------------------
__global__ __launch_bounds__(256) void knn_kernel(const float* __restrict__ pos,
                                                  int* __restrict__ idxout) {
  __shared__ float sd[8][32][16];
  __shared__ int   si[8][32][16];
  const int lane = threadIdx.x & 31;
  const int wv   = threadIdx.x >> 5;
  const int gp   = blockIdx.x * 8 + wv;       // global point 0..32767
  const int b    = gp >> 12;
  const int p    = gp & (PPTS - 1);
  const float* pb = pos + (size_t)b * PPTS * 3;
  const float px = pb[p * 3 + 0], py = pb[p * 3 + 1], pz = pb[p * 3 + 2];
  const float sqp = px * px + py * py + pz * pz;

  float dl[16];
  int   il[16];
#pragma unroll
  for (int t = 0; t < 16; ++t) { dl[t] = 3.0e38f; il[t] = -1; }

  for (int j = lane; j < PPTS; j += 32) {
    const float qx = pb[j * 3 + 0], qy = pb[j * 3 + 1], qz = pb[j * 3 + 2];
    float d2 = sqp + (qx * qx + qy * qy + qz * qz)
             - 2.0f * (px * qx + py * qy + pz * qz);
    if (j == p) d2 += 1.0e10f;                 // mask self (eye * 1e10)
    if (d2 < dl[15]) {
      // stable insert position: after equal distances (lower idx first)
      int ip = 0;
#pragma unroll
      for (int t = 0; t < 16; ++t) ip += (dl[t] <= d2) ? 1 : 0;
#pragma unroll
      for (int t = 15; t > 0; --t) {
        const bool sh = (t > ip);
        dl[t] = sh ? dl[t - 1] : dl[t];
        il[t] = sh ? il[t - 1] : il[t];
      }
#pragma unroll
      for (int t = 0; t < 16; ++t)
        if (t == ip) { dl[t] = d2; il[t] = j; }
    }
  }

  // stage per-lane sorted lists to LDS (same-wave DS ops stay in order)
#pragma unroll
  for (int t = 0; t < 16; ++t) { sd[wv][lane][t] = dl[t]; si[wv][lane][t] = il[t]; }

  int head = 0;
  for (int k = 0; k < KNN; ++k) {
    float d = sd[wv][lane][head];
    int   id = si[wv][lane][head];
    int   ln = lane;
#pragma unroll
    for (int off = 16; off > 0; off >>= 1) {
      const float od  = __shfl_xor(d, off, 32);
      const int   oid = __shfl_xor(id, off, 32);
      const int   oln = __shfl_xor(ln, off, 32);
      const bool take = (od < d) || (od == d && (unsigned)oid < (unsigned)id);
      d  = take ? od  : d;
      id = take ? oid : id;
      ln = take ? oln : ln;
    }
    if (lane == ln) head++;                    // winner advances its list
    if (lane == 0) idxout[(size_t)gp * KNN + k] = id;
  }
}

// ---------------------------------------------------------------------------
// GEMM  Y[M,DOUT] = X[M,DIN] @ W[DIN,DOUT] + bias  via V_WMMA_F32_16X16X4_F32
// Compile-time DIN/DOUT: no guards, no divergent loads. Each wave computes a
// 64x16 C strip (4 M-tiles sharing one B fragment per K-step -> 4 independent
// back-to-back WMMAs per B load). 4 waves/block cover 64 N columns.
// A 16x4 frag: lane&15 = M row, (lane>>4)*2 = K pair (contiguous -> b64 load).
// B 4x16 frag: lane&15 = N col, same K-pair striping.
// C 16x16 f32: 8 VGPRs, M = i + (lane>>4)*8, N = lane&15.
// ---------------------------------------------------------------------------
template <int DIN, int DOUT>
__global__ __launch_bounds__(128) void gemm_wmma_kernel(
    const float* __restrict__ X, const float* __restrict__ W,
    const float* __restrict__ bias, float* __restrict__ Y) {
  const int lane = threadIdx.x & 31;
  const int wv   = threadIdx.x >> 5;                 // 0..3
  const int m0   = blockIdx.x * 64;                  // 4 M-tiles per wave
  const int n0   = (blockIdx.y * 4 + wv) * 16;
  const int col  = lane & 15;                        // A row / B-C col
  const int kh   = (lane >> 4) << 1;                 // 0 or 2

  v8f acc[4] = {v8f{0.f}, v8f{0.f}, v8f{0.f}, v8f{0.f}};

  if constexpr (DIN == 3) {
    // Single K-step (K padded 3->4); lane-half select instead of exec branch.
    const bool hi = (kh != 0);
    const float w0 = W[0 * DOUT + n0 + col];
    const float w1 = W[1 * DOUT + n0 + col];
    const float w2 = W[2 * DOUT + n0 + col];
    v2f b;
    b.x = hi ? w2 : w0;
    b.y = hi ? 0.0f : w1;
#pragma unroll
    for (int mt = 0; mt < 4; ++mt) {
      const int row = m0 + mt * 16 + col;
      const float x0 = X[row * 3 + 0];
      const float x1 = X[row * 3 + 1];
      const float x2 = X[row * 3 + 2];
      v2f a;
      a.x = hi ? x2 : x0;
      a.y = hi ? 0.0f : x1;
      acc[mt] = __builtin_amdgcn_wmma_f32_16x16x4_f32(
          false, a, false, b, (short)0, acc[mt], false, false);
    }
  } else {
#pragma unroll 4
    for (int kk = 0; kk < DIN; kk += 4) {
      const int ka = kk + kh;
      v2f b;
      b.x = W[(size_t)(ka)     * DOUT + n0 + col];
      b.y = W[(size_t)(ka + 1) * DOUT + n0 + col];
#pragma unroll
      for (int mt = 0; mt < 4; ++mt) {
        const int row = m0 + mt * 16 + col;
        const v2f a = *(const v2f*)(X + (size_t)row * DIN + ka);  // b64 load
        acc[mt] = __builtin_amdgcn_wmma_f32_16x16x4_f32(
            false, a, false, b, (short)0, acc[mt], false, false);
      }
    }
  }

  const float bb     = bias[n0 + col];
  const int   rowOff = (lane >> 4) << 3;             // 0 or 8
#pragma unroll
  for (int mt = 0; mt < 4; ++mt)
#pragma unroll
    for (int i = 0; i < 8; ++i)
      Y[(size_t)(m0 + mt * 16 + rowOff + i) * DOUT + n0 + col] = acc[mt][i] + bb;
}

// ---------------------------------------------------------------------------
// Attention over 16 neighbors + skip + ReLU. One wave32 per point.
// ---------------------------------------------------------------------------
__global__ __launch_bounds__(256) void attn_kernel(
    const float* __restrict__ Q, const float* __restrict__ Kx,
    const float* __restrict__ V, const float* __restrict__ S,
    const int* __restrict__ idx, float* __restrict__ out, int d) {
  const int lane = threadIdx.x & 31;
  const int wv   = threadIdx.x >> 5;
  const int p    = blockIdx.x * 8 + wv;
  const int base = (p >> 12) << 12;                  // batch row offset

  const int* ipn = idx + (size_t)p * KNN;
  int nb[16];
#pragma unroll
  for (int k = 0; k < 16; ++k) nb[k] = base + ipn[k];

  const float* qp = Q + (size_t)p * d;
  float logit[16];
#pragma unroll
  for (int k = 0; k < 16; ++k) {
    const float* kp = Kx + (size_t)nb[k] * d;
    float s = 0.0f;
    for (int c = lane; c < d; c += 32) s += qp[c] * kp[c];
    logit[k] = wave_sum32(s);
  }

  const float scale = rsqrtf((float)d);
  float m = -3.0e38f;
#pragma unroll
  for (int k = 0; k < 16; ++k) { logit[k] *= scale; m = fmaxf(m, logit[k]); }
  float a[16], sum = 0.0f;
#pragma unroll
  for (int k = 0; k < 16; ++k) { a[k] = expf(logit[k] - m); sum += a[k]; }
  const float inv = 1.0f / sum;
#pragma unroll
  for (int k = 0; k < 16; ++k) a[k] *= inv;

  for (int c = lane; c < d; c += 32) {
    float acc = S[(size_t)p * d + c];                // skip = x@Ws + bs
#pragma unroll
    for (int k = 0; k < 16; ++k) acc += a[k] * V[(size_t)nb[k] * d + c];
    out[(size_t)p * d + c] = fmaxf(acc, 0.0f);       // ReLU
  }
}

// ---------------------------------------------------------------------------
// Global max pool over points: X[B*P,256] -> G[B,256]
// ---------------------------------------------------------------------------
__global__ __launch_bounds__(256) void maxpool_kernel(const float* __restrict__ X,
                                                      float* __restrict__ G) {
  const int c = threadIdx.x;
  const int b = blockIdx.x;
  const float* xb = X + (size_t)b * PPTS * 256;
  float m = -3.0e38f;
  for (int p = 0; p < PPTS; ++p) m = fmaxf(m, xb[(size_t)p * 256 + c]);
  G[b * 256 + c] = m;
}

// ---------------------------------------------------------------------------
// MLP head + log_softmax: one block per batch row.
// ---------------------------------------------------------------------------
__global__ __launch_bounds__(512) void head_kernel(
    const float* __restrict__ G,
    const float* __restrict__ w1, const float* __restrict__ b1,
    const float* __restrict__ w2, const float* __restrict__ b2,
    const float* __restrict__ w3, const float* __restrict__ b3,
    float* __restrict__ out) {
  __shared__ float sg[256], sh1[512], sh2[256], sl[10], sred[2];
  const int t = threadIdx.x;
  const int b = blockIdx.x;
  if (t < 256) sg[t] = G[b * 256 + t];
  __syncthreads();
  {  // fc1: 256 -> 512, ReLU
    float acc = b1[t];
    for (int c = 0; c < 256; ++c) acc += sg[c] * w1[c * 512 + t];
    sh1[t] = fmaxf(acc, 0.0f);
  }
  __syncthreads();
  if (t < 256) {  // fc2: 512 -> 256, ReLU
    float acc = b2[t];
    for (int c = 0; c < 512; ++c) acc += sh1[c] * w2[c * 256 + t];
    sh2[t] = fmaxf(acc, 0.0f);
  }
  __syncthreads();
  if (t < 10) {   // fc3: 256 -> 10
    float acc = b3[t];
    for (int c = 0; c < 256; ++c) acc += sh2[c] * w3[c * 10 + t];
    sl[t] = acc;
  }
  __syncthreads();
  if (t == 0) {
    float m = -3.0e38f;
    for (int k = 0; k < 10; ++k) m = fmaxf(m, sl[k]);
    float s = 0.0f;
    for (int k = 0; k < 10; ++k) s += expf(sl[k] - m);
    sred[0] = m; sred[1] = logf(s);
  }
  __syncthreads();
  if (t < 10) out[b * 10 + t] = sl[t] - sred[0] - sred[1];
}

// ---------------------------------------------------------------------------
extern "C" void kernel_launch(void* const* d_in, const int* in_sizes, int n_in,
                              void* d_out, int out_size, void* d_ws, size_t ws_size,
                              hipStream_t stream) {
  (void)in_sizes; (void)n_in; (void)out_size; (void)ws_size;
  const float* pos = (const float*)d_in[0];
  const float* W[3][4];
  const float* Bv[3][4];
  int ii = 1;
  for (int l = 0; l < 3; ++l)
    for (int nm = 0; nm < 4; ++nm) {
      W[l][nm]  = (const float*)d_in[ii++];
      Bv[l][nm] = (const float*)d_in[ii++];
    }
  const float* fc1w = (const float*)d_in[25]; const float* fc1b = (const float*)d_in[26];
  const float* fc2w = (const float*)d_in[27]; const float* fc2b = (const float*)d_in[28];
  const float* fc3w = (const float*)d_in[29]; const float* fc3b = (const float*)d_in[30];

  // Workspace layout (256B aligned)
  char*  ws  = (char*)d_ws;
  size_t off = 0;
  auto alloc = [&](size_t bytes) -> void* {
    void* p = ws + off;
    off = (off + bytes + 255) & ~(size_t)255;
    return p;
  };
  int*   IDX = (int*)alloc((size_t)NPTS * KNN * sizeof(int));
  const size_t FB = (size_t)NPTS * 256 * sizeof(float);
  float* Qb = (float*)alloc(FB);
  float* Kb = (float*)alloc(FB);
  float* Vb = (float*)alloc(FB);
  float* Sb = (float*)alloc(FB);
  float* XA = (float*)alloc(FB);
  float* XB = (float*)alloc(FB);
  float* G  = (float*)alloc((size_t)BD * 256 * sizeof(float));

  // 1) KNN
  knn_kernel<<<NPTS / 8, 256, 0, stream>>>(pos, IDX);

  // 2) layer 1: din=3 -> dout=64   (grid.x = Mtiles/4, grid.y = DOUT/64)
  {
    dim3 grid(NPTS / 64, 64 / 64);
    gemm_wmma_kernel<3, 64><<<grid, 128, 0, stream>>>(pos, W[0][0], Bv[0][0], Qb);
    gemm_wmma_kernel<3, 64><<<grid, 128, 0, stream>>>(pos, W[0][1], Bv[0][1], Kb);
    gemm_wmma_kernel<3, 64><<<grid, 128, 0, stream>>>(pos, W[0][2], Bv[0][2], Vb);
    gemm_wmma_kernel<3, 64><<<grid, 128, 0, stream>>>(pos, W[0][3], Bv[0][3], Sb);
    attn_kernel<<<NPTS / 8, 256, 0, stream>>>(Qb, Kb, Vb, Sb, IDX, XA, 64);
  }
  // 3) layer 2: 64 -> 128
  {
    dim3 grid(NPTS / 64, 128 / 64);
    gemm_wmma_kernel<64, 128><<<grid, 128, 0, stream>>>(XA, W[1][0], Bv[1][0], Qb);
    gemm_wmma_kernel<64, 128><<<grid, 128, 0, stream>>>(XA, W[1][1], Bv[1][1], Kb);
    gemm_wmma_kernel<64, 128><<<grid, 128, 0, stream>>>(XA, W[1][2], Bv[1][2], Vb);
    gemm_wmma_kernel<64, 128><<<grid, 128, 0, stream>>>(XA, W[1][3], Bv[1][3], Sb);
    attn_kernel<<<NPTS / 8, 256, 0, stream>>>(Qb, Kb, Vb, Sb, IDX, XB, 128);
  }
  // 4) layer 3: 128 -> 256
  {
    dim3 grid(NPTS / 64, 256 / 64);
    gemm_wmma_kernel<128, 256><<<grid, 128, 0, stream>>>(XB, W[2][0], Bv[2][0], Qb);
    gemm_wmma_kernel<128, 256><<<grid, 128, 0, stream>>>(XB, W[2][1], Bv[2][1], Kb);
    gemm_wmma_kernel<128, 256><<<grid, 128, 0, stream>>>(XB, W[2][2], Bv[2][2], Vb);
    gemm_wmma_kernel<128, 256><<<grid, 128, 0, stream>>>(XB, W[2][3], Bv[2][3], Sb);
    attn_kernel<<<NPTS / 8, 256, 0, stream>>>(Qb, Kb, Vb, Sb, IDX, XA, 256);
  }

  // 5) pool + head
  maxpool_kernel<<<BD, 256, 0, stream>>>(XA, G);
  head_kernel<<<BD, 512, 0, stream>>>(G, fc1w, fc1b, fc2w, fc2b, fc3w, fc3b,
                                      (float*)d_out);
}